// GuidedAttentionL1Loss_69183333204394
// MI455X (gfx1250) — compile-verified
//
#include <hip/hip_runtime.h>
#include <hip/hip_bf16.h>
#include <math.h>

#define B_BAGS 4096
#define ALPHA 1e-3f
#define BETA 0.1f
#define MIN_STDF 1.0f
#define MAX_STDF 1000.0f
#define MAX_LEN 3072
#define INV_SQRT_2PI 0.3989422804014327f

typedef float v2f  __attribute__((ext_vector_type(2)));
typedef float v8f  __attribute__((ext_vector_type(8)));
typedef unsigned int u32x4 __attribute__((ext_vector_type(4)));
typedef int i32x4 __attribute__((ext_vector_type(4)));
typedef int i32x8 __attribute__((ext_vector_type(8)));

__device__ __forceinline__ float wave_sum(float v) {
#pragma unroll
  for (int off = 16; off > 0; off >>= 1) v += __shfl_down(v, off, 32);
  return v;
}

// 3-way block reduction with broadcast; sm must hold >= 24 floats, block = 256.
__device__ __forceinline__ void block_reduce3(float& a, float& b, float& c, float* sm) {
  __syncthreads();
  a = wave_sum(a); b = wave_sum(b); c = wave_sum(c);
  const int lane = threadIdx.x & 31, wid = threadIdx.x >> 5;
  if (lane == 0) { sm[wid] = a; sm[8 + wid] = b; sm[16 + wid] = c; }
  __syncthreads();
  if (wid == 0) {
    float x = (lane < 8) ? sm[lane] : 0.f;
    float y = (lane < 8) ? sm[8 + lane] : 0.f;
    float z = (lane < 8) ? sm[16 + lane] : 0.f;
    x = wave_sum(x); y = wave_sum(y); z = wave_sum(z);
    if (lane == 0) { sm[0] = x; sm[8] = y; sm[16] = z; }
  }
  __syncthreads();
  a = sm[0]; b = sm[8]; c = sm[16];
}

// ---------------- per-bag attention penalty ----------------
// One block per bag. Stage the whole segment into LDS with the Tensor Data
// Mover (wave 0 programs a 1-row D#; FORCED this round — no __has_builtin
// guard — so the compile outcome itself proves whether TDM lowers), then two
// LDS passes:
//   pass A: {sum y, sum x*y, sum y^2}  -> mean
//   pass B: {sum r, sum y*r, sum r^2}  -> sum((y - r/S)^2) in closed form.
__global__ __launch_bounds__(256) void bag_kernel(
    const float* __restrict__ y, const int* __restrict__ starts,
    const int* __restrict__ lengths, const int* __restrict__ labels,
    float* __restrict__ bag_out) {
  __shared__ float lds_y[MAX_LEN];
  __shared__ float sred[24];
  const int b = blockIdx.x;
  const int t = threadIdx.x;
  const int start = starts[b];
  const int len = lengths[b];
  const int lab = labels[b];
  const float* ysrc = y + start;

#if defined(__gfx1250__)
  if ((t >> 5) == 0) {  // wave 0 only: TDM issues once per executing wave
    const unsigned long long ga = (unsigned long long)(uintptr_t)ysrc;
    const unsigned int lds_off = (unsigned int)(uintptr_t)(&lds_y[0]);
    u32x4 g0;
    g0[0] = 1u;                                   // count=1, gather off
    g0[1] = lds_off;                              // lds_addr
    g0[2] = (unsigned int)(ga & 0xFFFFFFFFull);   // global_addr lo
    g0[3] = ((unsigned int)(ga >> 32) & 0x01FFFFFFu) | 0x80000000u;  // hi | type=2
    i32x8 g1;
    g1[0] = 0x00020000;                 // workgroup_mask=0, data_size=4B
    g1[1] = (len & 0xFFFF) << 16;       // tensor_dim0[15:0]
    g1[2] = ((len >> 16) & 0xFFFF) | (1 << 16);  // tensor_dim0 hi | tensor_dim1=1
    g1[3] = (len & 0xFFFF) << 16;       // tile_dim0
    g1[4] = 1;                          // tile_dim1=1, tile_dim2=0
    g1[5] = len;                        // tensor_dim0_stride lo
    g1[6] = 0;
    g1[7] = 0;
    i32x4 gz = {0, 0, 0, 0};
#if __clang_major__ >= 23
    i32x8 gz8 = {0, 0, 0, 0, 0, 0, 0, 0};
    __builtin_amdgcn_tensor_load_to_lds(g0, g1, gz, gz, gz8, 0);
#else
    __builtin_amdgcn_tensor_load_to_lds(g0, g1, gz, gz, 0);
#endif
    __builtin_amdgcn_s_wait_tensorcnt(0);
  }
  __syncthreads();
#else
  for (int i = t; i < len; i += 256) lds_y[i] = ysrc[i];
  __syncthreads();
#endif

  const float inv_len = 1.0f / (float)len;
  float s_y = 0.f, s_xy = 0.f, s_y2 = 0.f;
  for (int i = t; i < len; i += 256) {
    const float yv = lds_y[i];
    const float x = (float)(i + 1) * inv_len;
    s_y += yv;
    s_xy += x * yv;
    s_y2 += yv * yv;
  }
  block_reduce3(s_y, s_xy, s_y2, sred);

  const float mean = s_xy / s_y;
  const float istd = (lab == 1 ? MIN_STDF : MAX_STDF) * inv_len;
  const float inv_istd = 1.0f / istd;
  const float inv_norm = inv_istd * INV_SQRT_2PI;

  float s_r = 0.f, s_yr = 0.f, s_r2 = 0.f;
  for (int i = t; i < len; i += 256) {
    const float x = (float)(i + 1) * inv_len;
    const float z = (x - mean) * inv_istd;
    const float rh = __expf(-0.5f * z * z) * inv_norm;
    const float yv = lds_y[i];
    s_r += rh;
    s_yr += yv * rh;
    s_r2 += rh * rh;
  }
  block_reduce3(s_r, s_yr, s_r2, sred);

  if (t == 0) {
    const float S = s_r + 1e-6f;
    const float invS = 1.0f / S;
    const float seg = s_y2 - 2.0f * s_yr * invS + s_r2 * invS * invS;
    bag_out[b] = seg * inv_len;
  }
}

// ---------------- prefix-sum of lengths + NLL ----------------
__global__ __launch_bounds__(1024) void scan_nll_kernel(
    const int* __restrict__ lengths, const float* __restrict__ logits,
    const int* __restrict__ labels, int* __restrict__ starts,
    float* __restrict__ nll_out) {
  __shared__ int sscan[1024];
  __shared__ float sred[32];
  const int t = threadIdx.x;
  int l[4];
#pragma unroll
  for (int j = 0; j < 4; ++j) l[j] = lengths[4 * t + j];
  sscan[t] = l[0] + l[1] + l[2] + l[3];
  __syncthreads();
  for (int off = 1; off < 1024; off <<= 1) {  // Hillis-Steele inclusive scan
    int add = (t >= off) ? sscan[t - off] : 0;
    __syncthreads();
    sscan[t] += add;
    __syncthreads();
  }
  const int excl = (t > 0) ? sscan[t - 1] : 0;
  starts[4 * t + 0] = excl;
  starts[4 * t + 1] = excl + l[0];
  starts[4 * t + 2] = excl + l[0] + l[1];
  starts[4 * t + 3] = excl + l[0] + l[1] + l[2];

  float nll = 0.f;  // 2-class log_softmax NLL
#pragma unroll
  for (int j = 0; j < 4; ++j) {
    const int row = 4 * t + j;
    const float a = logits[2 * row], b = logits[2 * row + 1];
    const int lab = labels[row];
    const float m = fmaxf(a, b);
    const float lse = m + __logf(__expf(a - m) + __expf(b - m));
    nll += lse - (lab ? b : a);
  }
  nll = wave_sum(nll);
  const int lane = t & 31, wid = t >> 5;
  if (lane == 0) sred[wid] = nll;
  __syncthreads();
  if (wid == 0) {
    float v = wave_sum(sred[lane]);  // exactly 32 waves
    if (lane == 0) nll_out[0] = v / (float)B_BAGS;
  }
}

// ---------------- L1(params) via WMMA abs-sum ----------------
// Block handles 4096 floats; each wave does 8 iterations of a 16x4 A-tile
// (64 f32) times a 4x16 ones B-tile, accumulating row sums in the f32
// accumulator. Sum(D) == 16 * sum(|A|), so divide by 16 at the end.
__global__ __launch_bounds__(256) void l1_kernel(const float* __restrict__ params,
                                                 float* __restrict__ pp) {
  __shared__ float sred[8];
  const int t = threadIdx.x;
  const int lane = t & 31, wid = t >> 5;
  const int base = blockIdx.x * 4096 + wid * 64 + lane * 2;
  float wsum;
#if defined(__gfx1250__) && __has_builtin(__builtin_amdgcn_wmma_f32_16x16x4_f32)
  v8f acc = {0.f, 0.f, 0.f, 0.f, 0.f, 0.f, 0.f, 0.f};
  v2f onesv = {1.0f, 1.0f};
#pragma unroll
  for (int it = 0; it < 8; ++it) {
    const float* p = params + base + it * 512;
    v2f a;
    a.x = fabsf(p[0]);
    a.y = fabsf(p[1]);
    acc = __builtin_amdgcn_wmma_f32_16x16x4_f32(
        /*neg_a=*/false, a, /*neg_b=*/false, onesv,
        /*c_mod=*/(short)0, acc, /*reuse_a=*/false, /*reuse_b=*/false);
  }
  float s = acc[0] + acc[1] + acc[2] + acc[3] + acc[4] + acc[5] + acc[6] + acc[7];
  wsum = wave_sum(s) * (1.0f / 16.0f);
#else
  float s = 0.f;
#pragma unroll
  for (int it = 0; it < 8; ++it) {
    const float* p = params + base + it * 512;
    s += fabsf(p[0]) + fabsf(p[1]);
  }
  wsum = wave_sum(s);
#endif
  if (lane == 0) sred[wid] = wsum;
  __syncthreads();
  if (t == 0) {
    float v = 0.f;
#pragma unroll
    for (int w = 0; w < 8; ++w) v += sred[w];
    pp[blockIdx.x] = v;
  }
}

// ---------------- finalize ----------------
__global__ __launch_bounds__(256) void finalize_kernel(
    const float* __restrict__ pp, const float* __restrict__ bag,
    const float* __restrict__ nllp, float* __restrict__ out) {
  __shared__ float sred[24];
  const int t = threadIdx.x;
  float ps = pp[t];  // 256 partials, one per thread
  float bs = 0.f;
  for (int i = t; i < B_BAGS; i += 256) bs += bag[i];
  float dummy = 0.f;
  block_reduce3(ps, bs, dummy, sred);
  if (t == 0) {
    const float nll = nllp[0];
    const float loss =
        nll + 0.5f * ALPHA * ps + 0.5f * BETA * (bs / (float)B_BAGS);
    out[0] = loss;
    out[1] = nll;
  }
}

extern "C" void kernel_launch(void* const* d_in, const int* in_sizes, int n_in,
                              void* d_out, int out_size, void* d_ws,
                              size_t ws_size, hipStream_t stream) {
  (void)in_sizes; (void)n_in; (void)out_size; (void)ws_size;
  const float* logits  = (const float*)d_in[0];
  const float* params  = (const float*)d_in[1];
  const float* attn    = (const float*)d_in[2];
  const int*   labels  = (const int*)d_in[3];
  const int*   lengths = (const int*)d_in[4];
  // d_in[5] = seg_ids: unused (starts are recomputed from lengths)

  char* w = (char*)d_ws;
  int*   starts = (int*)w;                // 4096 ints  @ 0
  float* nllp   = (float*)(w + 16384);    // 1 float
  float* pp     = (float*)(w + 16448);    // 256 floats
  float* bag    = (float*)(w + 20480);    // 4096 floats
  float* out    = (float*)d_out;

  scan_nll_kernel<<<1, 1024, 0, stream>>>(lengths, logits, labels, starts, nllp);
  l1_kernel<<<256, 256, 0, stream>>>(params, pp);
  bag_kernel<<<B_BAGS, 256, 0, stream>>>(attn, starts, lengths, labels, bag);
  finalize_kernel<<<1, 256, 0, stream>>>(pp, bag, nllp, out);
}